// HybridQuantumWaveletBottleneck_54924041781638
// MI455X (gfx1250) — compile-verified
//
#include <hip/hip_runtime.h>
#include <math.h>

// ---------------------------------------------------------------------------
// HybridQuantumWaveletBottleneck for MI455X (gfx1250, wave32, WMMA + TDM)
//
// Bandwidth-bound: ~0.94 GB minimum HBM traffic -> ~40us @ 23.3 TB/s.
// Channel-mix GEMM (17.2 GFLOP f32) via V_WMMA_F32_16X16X4_F32; fuse_W is
// staged into LDS by the Tensor Data Mover (TENSOR_LOAD_TO_LDS) with the
// TDM pad feature generating the conflict-free 132-float pitch.
// ---------------------------------------------------------------------------

#define NQ 8
#define NL 3
#define NB 32
#define NC 128
#define NH 128
#define NW 128
#define NSH 64
#define NSW 64
#define NFLAT (NC * NSH * NSW) // 524288

typedef __attribute__((ext_vector_type(2))) float v2f;
typedef __attribute__((ext_vector_type(8))) float v8f;
typedef __attribute__((ext_vector_type(4))) unsigned int v4u;
typedef __attribute__((ext_vector_type(4))) int v4i;
typedef __attribute__((ext_vector_type(8))) int v8i;

// ---------------------------------------------------------------------------
// zero the q_in accumulator (harness does not re-poison between replays)
// ---------------------------------------------------------------------------
__global__ void k_zero(float* __restrict__ p, int n) {
  int i = blockIdx.x * 256 + threadIdx.x;
  if (i < n) p[i] = 0.f;
}

// ---------------------------------------------------------------------------
// Phase 1: fused Haar DWT + q_in projection.
// q_in[s,b,q] = sum_flat dwt_s(x)[b,flat] * q_in_W[s,q,flat]
// Block: 256 threads = (32 batches) x (8 qubit-waves). Weight loads are
// wave-uniform (scalar broadcast). x4 unroll: base is 1024-aligned and rows
// are 64 sites, so 4 consecutive sites never cross a row -> float4 loads.
// ---------------------------------------------------------------------------
__global__ __launch_bounds__(256) void k_phase1(const float* __restrict__ xg,
                                                const float* __restrict__ q_in_W,
                                                float* __restrict__ qin) {
  const int tid = threadIdx.x;
  const int tb = tid & 31;  // batch
  const int tq = tid >> 5;  // qubit (wave index)
  const int base = blockIdx.x * 1024; // 512 blocks x 1024 sites = NFLAT
  float a0 = 0.f, a1 = 0.f, a2 = 0.f, a3 = 0.f;
  for (int i = 0; i < 1024; i += 4) {
    const int flat = base + i;
    const int c = flat >> 12;
    const int h2 = (flat >> 6) & 63;
    const int w2 = flat & 63; // multiple of 4
    const size_t xoff = (((size_t)tb * NC + c) * NH + 2 * h2) * NW + 2 * w2;
    const float4 x0 = *(const float4*)(xg + xoff);          // row 2h2
    const float4 x1 = *(const float4*)(xg + xoff + 4);
    const float4 y0 = *(const float4*)(xg + xoff + NW);     // row 2h2+1
    const float4 y1 = *(const float4*)(xg + xoff + NW + 4);
    const size_t wb = (size_t)tq * NFLAT + flat;            // [s][q][flat]
    const float4 wll = *(const float4*)(q_in_W + wb);
    const float4 wlh = *(const float4*)(q_in_W + wb + (size_t)8 * NFLAT);
    const float4 whl = *(const float4*)(q_in_W + wb + (size_t)16 * NFLAT);
    const float4 whh = *(const float4*)(q_in_W + wb + (size_t)24 * NFLAT);
    const float ax[4] = {x0.x, x0.z, x1.x, x1.z};
    const float bx[4] = {x0.y, x0.w, x1.y, x1.w};
    const float cx[4] = {y0.x, y0.z, y1.x, y1.z};
    const float dx[4] = {y0.y, y0.w, y1.y, y1.w};
    const float wl0[4] = {wll.x, wll.y, wll.z, wll.w};
    const float wl1[4] = {wlh.x, wlh.y, wlh.z, wlh.w};
    const float wl2[4] = {whl.x, whl.y, whl.z, whl.w};
    const float wl3[4] = {whh.x, whh.y, whh.z, whh.w};
#pragma unroll
    for (int j = 0; j < 4; ++j) {
      const float sA = ax[j] + bx[j], dA = ax[j] - bx[j];
      const float sC = cx[j] + dx[j], dC = cx[j] - dx[j];
      a0 = fmaf(sA + sC, wl0[j], a0); // ll
      a1 = fmaf(sA - sC, wl1[j], a1); // lh
      a2 = fmaf(dA + dC, wl2[j], a2); // hl
      a3 = fmaf(dA - dC, wl3[j], a3); // hh
    }
  }
  const int qi = tb * 8 + tq;
  atomicAdd(&qin[0 * 256 + qi], 0.5f * a0);
  atomicAdd(&qin[1 * 256 + qi], 0.5f * a1);
  atomicAdd(&qin[2 * 256 + qi], 0.5f * a2);
  atomicAdd(&qin[3 * 256 + qi], 0.5f * a3);
}

// ---------------------------------------------------------------------------
// Phase 2: 8-qubit statevector circuit, one block per (subband, batch).
// Qubit q maps to state-index bit (7-q) (row-major reshape in the reference).
// ---------------------------------------------------------------------------
__device__ __forceinline__ float2 cadd(float2 a, float2 b) {
  return make_float2(a.x + b.x, a.y + b.y);
}
__device__ __forceinline__ float2 cmul(float2 a, float2 b) {
  return make_float2(a.x * b.x - a.y * b.y, a.x * b.y + a.y * b.x);
}

__global__ __launch_bounds__(128) void k_phase2(const float* __restrict__ qin,
                                                const float* __restrict__ q_in_b,
                                                const float* __restrict__ q_weights,
                                                float* __restrict__ ev) {
  __shared__ float2 st[256];
  __shared__ float pr[256];
  const int tid = threadIdx.x;
  const int s = blockIdx.x >> 5;
  const int b = blockIdx.x & 31;

  st[tid] = make_float2(0.f, 0.f);
  st[tid + 128] = make_float2(0.f, 0.f);
  __syncthreads();
  if (tid == 0) st[0] = make_float2(1.f, 0.f);
  __syncthreads();

  auto apply1q = [&](float2 U00, float2 U01, float2 U10, float2 U11, int q) {
    const int mask = 1 << (7 - q);
    const int low = tid & (mask - 1);
    const int i0 = ((tid & ~(mask - 1)) << 1) | low;
    const int i1 = i0 | mask;
    const float2 v0 = st[i0], v1 = st[i1];
    st[i0] = cadd(cmul(U00, v0), cmul(U01, v1));
    st[i1] = cadd(cmul(U10, v0), cmul(U11, v1));
    __syncthreads();
  };

  // RX embedding layer (per-batch angles)
  for (int q = 0; q < NQ; ++q) {
    const float th = 0.5f * (qin[s * 256 + b * 8 + q] + q_in_b[s * 8 + q]);
    const float ch = cosf(th), sh = sinf(th);
    apply1q(make_float2(ch, 0.f), make_float2(0.f, -sh),
            make_float2(0.f, -sh), make_float2(ch, 0.f), q);
  }
  // Entangling layers (shared Rot weights + CNOT ring)
  for (int l = 0; l < NL; ++l) {
    for (int q = 0; q < NQ; ++q) {
      const float* wq = q_weights + (((size_t)s * NL + l) * NQ + q) * 3;
      const float phi = wq[0], theta = wq[1], omega = wq[2];
      const float ct = cosf(0.5f * theta), sq = sinf(0.5f * theta);
      const float al = 0.5f * (phi + omega), be = 0.5f * (phi - omega);
      const float ca = cosf(al), sa = sinf(al), cb = cosf(be), sb = sinf(be);
      // ep = (ca,-sa), em = (cb,-sb)
      apply1q(make_float2(ca * ct, -sa * ct),  // ep*ct
              make_float2(-cb * sq, -sb * sq), // -conj(em)*st
              make_float2(cb * sq, -sb * sq),  // em*st
              make_float2(ca * ct, sa * ct),   // conj(ep)*ct
              q);
    }
    for (int q = 0; q < NQ; ++q) {
      const int cq = (q < NQ - 1) ? q : NQ - 1;
      const int tq = (q < NQ - 1) ? q + 1 : 0;
      const int cmask = 1 << (7 - cq);
      const int tmask = 1 << (7 - tq);
      const int low = tid & (tmask - 1);
      const int i0 = ((tid & ~(tmask - 1)) << 1) | low;
      if (i0 & cmask) { // control bit set: swap target pair
        const int i1 = i0 | tmask;
        const float2 t0 = st[i0];
        st[i0] = st[i1];
        st[i1] = t0;
      }
      __syncthreads();
    }
  }
  // Expectation values <Z_q>
  pr[tid] = st[tid].x * st[tid].x + st[tid].y * st[tid].y;
  pr[tid + 128] = st[tid + 128].x * st[tid + 128].x + st[tid + 128].y * st[tid + 128].y;
  __syncthreads();
  if (tid < 8) {
    const int mask = 1 << (7 - tid);
    float sum = 0.f;
    for (int i = 0; i < 256; ++i) sum += (i & mask) ? -pr[i] : pr[i];
    ev[s * 256 + b * 8 + tid] = sum;
  }
}

// ---------------------------------------------------------------------------
// Phase 3: q_out projection + IDWT + channel-mix GEMM (WMMA) + residual.
// Block: fixed batch b, fixed h2, 32 consecutive w2 sites.
//   GEMM: out[o, site] = sum_c fuse_W[o,c] * rec[pos][c, site]
//   A = fuse_W (M=o=128, K=c=128), B = rec planes (K=c, N=site=32), 4 pos.
// fuse_W is DMA'd into LDS by the TDM (pad feature -> 132-float pitch) while
// the waves build the rec planes from ev . q_out_W + q_out_b (q_out_W reads
// fully coalesced, q innermost). LDS pitches 132/40 -> conflict-free frags.
// ---------------------------------------------------------------------------
__global__ __launch_bounds__(256) void k_phase3(const float* __restrict__ xg,
                                                const float* __restrict__ q_out_W,
                                                const float* __restrict__ q_out_b,
                                                const float* __restrict__ fuse_W,
                                                const float* __restrict__ fuse_b,
                                                const float* __restrict__ ev,
                                                float* __restrict__ out) {
  __shared__ float Af[128 * 132];        // fuse_W, A-layout [o][c], pitch 132
  __shared__ float recB[4 * 128 * 40];   // rec planes, B-layout [pos][c][site]
  __shared__ float fb[128];

  const int tid = threadIdx.x;
  const int blk = blockIdx.x;        // 32 b * 64 h2 * 2 wgroups = 4096
  const int wg = blk & 1;
  const int h2 = (blk >> 1) & 63;
  const int b = blk >> 7;
  const int w2base = wg * 32;
  const int wave = tid >> 5;

  // --- TDM: async DMA fuse_W (128x128 f32, row stride 128) into Af with the
  // pad feature inserting 4 dwords every 128 dwords (-> pitch 132). Issued
  // once, by wave 0; tracked by TENSORcnt. Overlaps with rec-plane build.
  if (wave == 0) {
    const unsigned long long ga = (unsigned long long)(const void*)fuse_W;
    const unsigned lds_off = (unsigned)(unsigned long long)(const void*)&Af[0];
    const v4u g0 = {
        1u,                                            // count=1 (valid D#)
        lds_off,                                       // lds_addr
        (unsigned)(ga & 0xffffffffu),                  // global_addr[31:0]
        (unsigned)((ga >> 32) & 0x01ffffffu) | (2u << 30) // addr[56:32]|type=2
    };
    const v8i g1 = {
        (int)((2u << 16) |            // data_size = 4B
              (1u << 20) |            // pad_enable
              (6u << 22) |            // pad_interval: 128 dwords
              (3u << 25)),            // pad_amount: 4 dwords
        (int)(128u << 16),            // tensor_dim0[15:0] = 128
        (int)(128u << 16),            // tensor_dim1[15:0] = 128
        (int)(128u << 16),            // tile_dim0 = 128
        (int)128u,                    // tile_dim1 = 128
        (int)128u,                    // tensor_dim0_stride = 128
        0, 0
    };
    const v4i gz = {0, 0, 0, 0};
#if defined(__clang_major__) && (__clang_major__ >= 23)
    const v8i gz8 = {0, 0, 0, 0, 0, 0, 0, 0};
    __builtin_amdgcn_tensor_load_to_lds(g0, g1, gz, gz, gz8, 0);
#else
    __builtin_amdgcn_tensor_load_to_lds(g0, g1, gz, gz, 0);
#endif
  }
  if (tid < 128) fb[tid] = fuse_b[tid];

  // Expectation values for this batch (wave-uniform scalar loads)
  float evr[4][8];
#pragma unroll
  for (int s = 0; s < 4; ++s)
#pragma unroll
    for (int q = 0; q < 8; ++q) evr[s][q] = ev[s * 256 + b * 8 + q];

  // Build the 4 IDWT-position rec planes (overlapped with the TDM transfer)
  for (int it = 0; it < 16; ++it) {
    const int p = tid + it * 256; // 4096 (c, site) pairs
    const int c = p >> 5, j = p & 31;
    const int flat = c * 4096 + h2 * 64 + w2base + j;
    float prc[4];
#pragma unroll
    for (int s = 0; s < 4; ++s) {
      const float4* wp =
          (const float4*)(q_out_W + ((size_t)s * NFLAT + flat) * 8);
      const float4 w0 = wp[0], w1 = wp[1];
      float acc = q_out_b[(size_t)s * NFLAT + flat];
      acc += evr[s][0] * w0.x + evr[s][1] * w0.y + evr[s][2] * w0.z +
             evr[s][3] * w0.w;
      acc += evr[s][4] * w1.x + evr[s][5] * w1.y + evr[s][6] * w1.z +
             evr[s][7] * w1.w;
      prc[s] = acc;
    }
    // idwt butterfly (a,b,c,d positions), 0.5 factor folded in here
    recB[(0 * 128 + c) * 40 + j] = 0.5f * (prc[0] + prc[1] + prc[2] + prc[3]);
    recB[(1 * 128 + c) * 40 + j] = 0.5f * (prc[0] + prc[1] - prc[2] - prc[3]);
    recB[(2 * 128 + c) * 40 + j] = 0.5f * (prc[0] - prc[1] + prc[2] - prc[3]);
    recB[(3 * 128 + c) * 40 + j] = 0.5f * (prc[0] - prc[1] - prc[2] + prc[3]);
  }
  // Fence the TDM transfer (wave 0 holds the TENSORcnt), then barrier.
  if (wave == 0) __builtin_amdgcn_s_wait_tensorcnt(0);
  __syncthreads();

  const int lane = tid & 31;
  const int lhalf = lane >> 4; // K split across lane halves for f32 WMMA
  const int l15 = lane & 15;

  // 64 tiles: 4 pos x 8 tm(o) x 2 tn(site); 8 tiles per wave
  for (int t = wave; t < 64; t += 8) {
    const int pos = t >> 4;
    const int tm = (t >> 1) & 7;
    const int tn = t & 1;
    const int arow = tm * 16 + l15;
    const int bn = tn * 16 + l15;
    v8f acc = {};
#pragma unroll 4
    for (int k0 = 0; k0 < 32; ++k0) {
      const int k = k0 * 4 + lhalf * 2;
      v2f afr, bfr;
      afr[0] = Af[arow * 132 + k];
      afr[1] = Af[arow * 132 + k + 1];
      bfr[0] = recB[(pos * 128 + k) * 40 + bn];
      bfr[1] = recB[(pos * 128 + k + 1) * 40 + bn];
      acc = __builtin_amdgcn_wmma_f32_16x16x4_f32(false, afr, false, bfr,
                                                  (short)0, acc, false, false);
    }
    // Writeout: out = x + fuse_b + gemm. Non-temporal to protect L2-resident
    // q_out_W from the streaming x/out traffic.
    const int hp = pos >> 1, wpar = pos & 1;
    const int h = 2 * h2 + hp;
#pragma unroll
    for (int r = 0; r < 8; ++r) {
      const int o = tm * 16 + r + lhalf * 8; // C/D layout: M = r (+8 hi half)
      const int w2 = w2base + tn * 16 + l15;
      const int w = 2 * w2 + wpar;
      const size_t idx = (((size_t)b * NC + o) * NH + h) * NW + w;
      const float xr = __builtin_nontemporal_load(&xg[idx]);
      __builtin_nontemporal_store(xr + fb[o] + acc[r], &out[idx]);
    }
  }
}

// ---------------------------------------------------------------------------
extern "C" void kernel_launch(void* const* d_in, const int* in_sizes, int n_in,
                              void* d_out, int out_size, void* d_ws,
                              size_t ws_size, hipStream_t stream) {
  const float* x = (const float*)d_in[0];
  const float* q_in_W = (const float*)d_in[1];
  const float* q_in_b = (const float*)d_in[2];
  const float* q_weights = (const float*)d_in[3];
  const float* q_out_W = (const float*)d_in[4];
  const float* q_out_b = (const float*)d_in[5];
  const float* fuse_W = (const float*)d_in[6];
  const float* fuse_b = (const float*)d_in[7];
  float* out = (float*)d_out;

  float* qin = (float*)d_ws;  // [4][32][8] accumulators
  float* ev = qin + 1024;     // [4][32][8] expectation values

  k_zero<<<4, 256, 0, stream>>>(qin, 1024);
  k_phase1<<<512, 256, 0, stream>>>(x, q_in_W, qin);
  k_phase2<<<128, 128, 0, stream>>>(qin, q_in_b, q_weights, ev);
  k_phase3<<<4096, 256, 0, stream>>>(x, q_out_W, q_out_b, fuse_W, fuse_b, ev,
                                     out);
}